// gauss_attention_24713241822145
// MI455X (gfx1250) — compile-verified
//
#include <hip/hip_runtime.h>

typedef float v2f __attribute__((ext_vector_type(2)));
typedef float v8f __attribute__((ext_vector_type(8)));

namespace {
constexpr int kN = 262144;
constexpr int kS = 9;
constexpr int kE = 32;
constexpr int kHD = 16;
constexpr float kSigma = 10.0f;
constexpr int kStride = 36;                  // padded LDS row stride (floats): bank-conflict-free, 16B aligned
constexpr int kWaves = 4;                    // waves per block
constexpr int kWB = 16;                      // batches per wave (= WMMA M)
constexpr int kBlockBatches = kWaves * kWB;  // 64
}

// Coalesced 16x32 f32 tile load: 4 rounds, 8 lanes x float4 per row (128B/row).
__device__ __forceinline__ void stage16x32(const float* __restrict__ src, int rowStride,
                                           float* __restrict__ dst, int lane) {
#pragma unroll
  for (int rr = 0; rr < 4; ++rr) {
    const int row = rr * 4 + (lane >> 3);
    const int col = (lane & 7) * 4;
    const float4 v = *reinterpret_cast<const float4*>(src + (size_t)row * rowStride + col);
    float* d = dst + row * kStride + col;
    d[0] = v.x; d[1] = v.y; d[2] = v.z; d[3] = v.w;
  }
}

// OUT(16x32) = X(16x32) @ W^T + bias  via V_WMMA_F32_16X16X4_F32 (8 K-steps, 2 N-tiles).
// W is [32 feat][32 e] row-major with kStride padding; D[m][f] = sum_e X[m][e]*W[f][e].
// Safe to call with OUT == X: all lanes issue every fragment load before any store (lockstep wave).
__device__ __forceinline__ void project16(const float* __restrict__ X,
                                          const float* __restrict__ W,
                                          const float* __restrict__ bias,
                                          float* __restrict__ OUT, int lane) {
  const int m = lane & 15;            // A: row M ; B: col N
  const int kh = (lane >> 4) << 1;    // lanes 16-31 hold K+2,K+3
  v8f c0 = {0.f, 0.f, 0.f, 0.f, 0.f, 0.f, 0.f, 0.f};
  v8f c1 = {0.f, 0.f, 0.f, 0.f, 0.f, 0.f, 0.f, 0.f};
#pragma unroll
  for (int k0 = 0; k0 < kE; k0 += 4) {
    v2f a, b0, b1;
    a[0]  = X[m * kStride + k0 + kh];
    a[1]  = X[m * kStride + k0 + kh + 1];
    b0[0] = W[m * kStride + k0 + kh];             // B[k][n] = W[n][k], n-tile 0
    b0[1] = W[m * kStride + k0 + kh + 1];
    b1[0] = W[(16 + m) * kStride + k0 + kh];      // n-tile 1
    b1[1] = W[(16 + m) * kStride + k0 + kh + 1];
    c0 = __builtin_amdgcn_wmma_f32_16x16x4_f32(false, a, false, b0, (short)0, c0, false, false);
    c1 = __builtin_amdgcn_wmma_f32_16x16x4_f32(false, a, false, b1, (short)0, c1, false, false);
  }
  const float bv0 = bias[m];
  const float bv1 = bias[16 + m];
  const int rb = (lane >> 4) * 8;     // D: vgpr r -> row r (+8 for upper lanes)
#pragma unroll
  for (int r = 0; r < 8; ++r) {
    OUT[(rb + r) * kStride + m]      = c0[r] + bv0;
    OUT[(rb + r) * kStride + 16 + m] = c1[r] + bv1;
  }
}

__global__ __launch_bounds__(kWaves * 32) void gauss_attention_24713241822145_kernel(
    const float* __restrict__ query, const float* __restrict__ key,
    const float* __restrict__ value, const float* __restrict__ gauss,
    const float* __restrict__ in_proj_weight, const float* __restrict__ in_proj_bias,
    const float* __restrict__ out_proj_weight, const float* __restrict__ out_proj_bias,
    float* __restrict__ out) {
  __shared__ float sW[4][kE * kStride];             // Wq, Wk, Wv, Wo (padded)
  __shared__ float sBuf[kWaves][3][kWB * kStride];  // per-wave: A, B, Q/O tiles

  const int tid = threadIdx.x;
  const int wave = tid >> 5;
  const int lane = tid & 31;

  for (int i = tid; i < 3 * kE * kE; i += kWaves * 32) {
    const int r = i >> 5, c = i & 31;
    sW[r >> 5][(r & 31) * kStride + c] = in_proj_weight[i];
  }
  for (int i = tid; i < kE * kE; i += kWaves * 32) {
    sW[3][(i >> 5) * kStride + (i & 31)] = out_proj_weight[i];
  }
  __syncthreads();

  const size_t base = (size_t)blockIdx.x * kBlockBatches + (size_t)wave * kWB;
  float* bufA = sBuf[wave][0];
  float* bufB = sBuf[wave][1];
  float* bufQ = sBuf[wave][2];

  // ---- Q projection (T = 1) ----
  stage16x32(query + base * kE, kE, bufA, lane);
  __syncthreads();
  project16(bufA, sW[0], in_proj_bias, bufQ, lane);
  __syncthreads();

  // One (batch, head) per lane: b = lane&15, h = lane>>4.
  const int b = lane & 15;
  const int h = lane >> 4;
  float qr[kHD];
#pragma unroll
  for (int d = 0; d < kHD; ++d)
    qr[d] = bufQ[b * kStride + h * kHD + d] * 0.25f;  // 1/sqrt(HD)

  // Online (flash) softmax state per lane.
  float mrun = -__builtin_inff();
  float lrun = 0.0f;
  float acc[kHD];
#pragma unroll
  for (int d = 0; d < kHD; ++d) acc[d] = 0.0f;

  for (int s = 0; s < kS; ++s) {
    stage16x32(key   + (base * kS + (size_t)s) * kE, kS * kE, bufA, lane);
    stage16x32(value + (base * kS + (size_t)s) * kE, kS * kE, bufB, lane);
    __syncthreads();
    project16(bufA, sW[1], in_proj_bias + kE,     bufA, lane);  // K tile (in place)
    project16(bufB, sW[2], in_proj_bias + 2 * kE, bufB, lane);  // V tile (in place)
    __syncthreads();

    float sc = 0.0f;
#pragma unroll
    for (int d = 0; d < kHD; ++d)
      sc += qr[d] * bufA[b * kStride + h * kHD + d];
    const float g = gauss[(base + b) * kS + s];
    sc *= __expf(-kSigma * g);          // gaussian multiplier applied pre-softmax

    const float nm = fmaxf(mrun, sc);
    const float corr = __expf(mrun - nm);
    const float p = __expf(sc - nm);
    lrun = lrun * corr + p;
#pragma unroll
    for (int d = 0; d < kHD; ++d)
      acc[d] = acc[d] * corr + p * bufB[b * kStride + h * kHD + d];
    mrun = nm;
    __syncthreads();
  }

  const float inv = 1.0f / lrun;
#pragma unroll
  for (int d = 0; d < kHD; ++d)
    bufQ[b * kStride + h * kHD + d] = acc[d] * inv;  // o_flat[b][h*16+d]
  __syncthreads();

  // ---- output projection ----
  project16(bufQ, sW[3], out_proj_bias, bufA, lane);
  __syncthreads();

#pragma unroll
  for (int rr = 0; rr < 4; ++rr) {
    const int row = rr * 4 + (lane >> 3);
    const int col = (lane & 7) * 4;
    float4 v;
    v.x = bufA[row * kStride + col + 0];
    v.y = bufA[row * kStride + col + 1];
    v.z = bufA[row * kStride + col + 2];
    v.w = bufA[row * kStride + col + 3];
    *reinterpret_cast<float4*>(out + (base + row) * kE + col) = v;
  }
}

extern "C" void kernel_launch(void* const* d_in, const int* in_sizes, int n_in,
                              void* d_out, int out_size, void* d_ws, size_t ws_size,
                              hipStream_t stream) {
  (void)in_sizes; (void)n_in; (void)out_size; (void)d_ws; (void)ws_size;
  const float* query  = (const float*)d_in[0];
  const float* key    = (const float*)d_in[1];
  const float* value  = (const float*)d_in[2];
  const float* gauss  = (const float*)d_in[3];
  const float* ipw    = (const float*)d_in[4];
  const float* ipb    = (const float*)d_in[5];
  const float* opw    = (const float*)d_in[6];
  const float* opb    = (const float*)d_in[7];
  float* out = (float*)d_out;

  dim3 grid(kN / kBlockBatches);   // 4096
  dim3 block(kWaves * 32);         // 128 threads = 4 waves
  hipLaunchKernelGGL(gauss_attention_24713241822145_kernel, grid, block, 0, stream,
                     query, key, value, gauss, ipw, ipb, opw, opb, out);
}